// Dash4d_63754494542257
// MI455X (gfx1250) — compile-verified
//
#include <hip/hip_runtime.h>
#include <cmath>

#define HG_BOUND 1.6f

// Prefer CDNA5 async global->LDS staging when the toolchain exposes it.
#if defined(__has_builtin)
#  if __has_builtin(__builtin_amdgcn_global_load_async_to_lds_b32) && \
      __has_builtin(__builtin_amdgcn_s_wait_asynccnt)
#    define HG_ASYNC_STAGE 1
#  endif
#endif
#ifndef HG_ASYNC_STAGE
#  define HG_ASYNC_STAGE 0
#endif

#define HG_AS_GLOBAL __attribute__((address_space(1)))
#define HG_AS_LDS    __attribute__((address_space(3)))

// ---------------------------------------------------------------------------
// Per-level metadata for a hash-grid encoding level. 48 bytes, 12 dwords.
// ---------------------------------------------------------------------------
struct LvlMeta {
    float    scale[4];   // res[d] as float (0 for unused dims)
    unsigned stride[4];  // dense strides: cumprod(res+1)
    unsigned offset;     // entry offset of this level in the table
    unsigned mask;       // size-1 (size == 2^19 when hashed, pow2)
    unsigned dense;      // 1 = dense addressing, 0 = hashed
    unsigned pad;
};

// ---------------------------------------------------------------------------
// Kernel: one thread per (point, level).
//   D      : input dimensionality (3 or 4)
//   NLEV   : number of levels (16 or 32), power of two
//   LSHIFT : log2(NLEV)
// Metadata is staged into LDS once per block (async global->LDS on the
// ASYNCcnt path when available, DS pipe otherwise), keeping the vmem pipe
// dedicated to the random table gathers that bound this kernel. Each thread
// issues 2^D independent global_load_b64 gathers (deep MLP under LOADcnt).
// ---------------------------------------------------------------------------
template <int D, int NLEV, int LSHIFT>
__global__ __launch_bounds__(256)
void hashgrid_encode(const float4* __restrict__ xyzt,
                     const float2* __restrict__ table,
                     const LvlMeta* __restrict__ gmeta,
                     float* __restrict__ out,
                     int npts)
{
    __shared__ LvlMeta smeta[NLEV];
    constexpr int NDW = NLEV * (int)(sizeof(LvlMeta) / 4);  // 192 or 384

    {
        unsigned*       sdw = (unsigned*)smeta;
        const unsigned* gdw = (const unsigned*)gmeta;
        const int t0 = (int)threadIdx.x;
#if HG_ASYNC_STAGE
        if (t0 < NDW)
            __builtin_amdgcn_global_load_async_to_lds_b32(
                (HG_AS_GLOBAL int*)(gdw + t0),
                (HG_AS_LDS int*)(sdw + t0), 0, 0);
        if (NDW > 256) {
            const int t1 = t0 + 256;
            if (t1 < NDW)
                __builtin_amdgcn_global_load_async_to_lds_b32(
                    (HG_AS_GLOBAL int*)(gdw + t1),
                    (HG_AS_LDS int*)(sdw + t1), 0, 0);
        }
        __builtin_amdgcn_s_wait_asynccnt(0);
#else
        if (t0 < NDW) sdw[t0] = gdw[t0];
        if (NDW > 256) {
            const int t1 = t0 + 256;
            if (t1 < NDW) sdw[t1] = gdw[t1];
        }
#endif
    }
    __syncthreads();

    const int tid = blockIdx.x * blockDim.x + threadIdx.x;
    const int i   = tid >> LSHIFT;
    if (i >= npts) return;
    const int l   = tid & (NLEV - 1);

    // One b128 load per thread; wave-uniform within NLEV-lane groups -> L0 hit.
    const float4 p4 = xyzt[i];
    const float  xin[4] = {p4.x, p4.y, p4.z, p4.w};

    const LvlMeta lv = smeta[l];   // 3x ds_load_b128

    float    frac[4];
    unsigned p0[4];
#pragma unroll
    for (int d = 0; d < D; ++d) {
        float v = (xin[d] + HG_BOUND) * (1.0f / (2.0f * HG_BOUND));
        v = fminf(fmaxf(v, 0.0f), (float)(1.0 - 1e-6));
        const float pos = v * lv.scale[d];
        float pf = floorf(pos);
        pf = fminf(fmaxf(pf, 0.0f), lv.scale[d] - 1.0f);
        frac[d] = pos - pf;
        p0[d]   = (unsigned)pf;
    }

    const unsigned PR[4] = {1u, 2654435761u, 805459861u, 3674653429u};
    unsigned dbase = 0;
    unsigned h0[4], h1[4];
#pragma unroll
    for (int d = 0; d < D; ++d) {
        dbase += p0[d] * lv.stride[d];
        h0[d] = p0[d] * PR[d];
        h1[d] = h0[d] + PR[d];
    }
    const bool dense = (lv.dense != 0);

    float acc0 = 0.0f, acc1 = 0.0f;
#pragma unroll
    for (int c = 0; c < (1 << D); ++c) {
        float    w    = 1.0f;
        unsigned lind = dbase;
        unsigned h    = 0;
#pragma unroll
        for (int d = 0; d < D; ++d) {
            if ((c >> d) & 1) { w *= frac[d];          lind += lv.stride[d]; h ^= h1[d]; }
            else              { w *= (1.0f - frac[d]);                        h ^= h0[d]; }
        }
        const unsigned lin = dense ? lind : (h & lv.mask);
        const float2   t   = table[lv.offset + lin];   // global_load_b64 gather
        acc0 = fmaf(w, t.x, acc0);
        acc1 = fmaf(w, t.y, acc1);
    }

    // Coalesced: consecutive lanes (same point, consecutive levels) write
    // consecutive float2 slots of the point's output row.
    float2* o = (float2*)(out + (size_t)i * (2 * NLEV));
    o[l] = make_float2(acc0, acc1);
}

// ---------------------------------------------------------------------------
// Host-side replication of the reference build_meta() in double precision.
// ---------------------------------------------------------------------------
static void build_meta_host(LvlMeta* out, int D, int nlev,
                            const double* baseIn, int nbase,
                            const double* desIn, int ndes,
                            int log2_hashmap_size)
{
    double base[4], des[4];
    for (int d = 0; d < D; ++d) {
        base[d] = baseIn[d < nbase ? d : nbase - 1];
        des[d]  = desIn [d < ndes  ? d : ndes  - 1];
    }
    double growth[4];
    const double denom = (nlev - 1 > 1) ? (double)(nlev - 1) : 1.0;
    for (int d = 0; d < D; ++d)
        growth[d] = exp2(log2(des[d] / base[d]) / denom);

    const unsigned long long hmax = 1ull << log2_hashmap_size;
    unsigned long long offset = 0;
    for (int l = 0; l < nlev; ++l) {
        long long res[4];
        unsigned long long full = 1;
        for (int d = 0; d < D; ++d) {
            res[d] = (long long)ceil(base[d] * pow(growth[d], (double)l));
            full  *= (unsigned long long)(res[d] + 1);
        }
        const unsigned long long size = (full < hmax) ? full : hmax;

        LvlMeta& m = out[l];
        for (int d = 0; d < 4; ++d) { m.scale[d] = 0.0f; m.stride[d] = 0u; }
        unsigned long long s = 1;
        for (int d = 0; d < D; ++d) {
            m.scale[d]  = (float)res[d];
            m.stride[d] = (unsigned)s;
            s *= (unsigned long long)(res[d] + 1);
        }
        m.offset = (unsigned)offset;
        m.mask   = (unsigned)(size - 1);
        m.dense  = (size == full) ? 1u : 0u;
        m.pad    = 0u;
        offset += size;
    }
}

// Host-persistent meta buffers: refilled with identical values on every call
// (deterministic); they must outlive graph capture since the H2D memcpy node
// re-reads them on each replay.
static LvlMeta g_meta3[16];
static LvlMeta g_meta4[32];

extern "C" void kernel_launch(void* const* d_in, const int* in_sizes, int n_in,
                              void* d_out, int out_size, void* d_ws, size_t ws_size,
                              hipStream_t stream)
{
    const float*  xyzt = (const float*)d_in[0];
    const float2* t3   = (const float2*)d_in[1];
    const float2* t4   = (const float2*)d_in[2];
    const int     n    = in_sizes[0] / 4;

    float* out3 = (float*)d_out;
    float* out4 = out3 + (size_t)n * 32;

    // META3: D=3, 16 levels, base 16 -> desired 2048, log2T=19
    const double base3[1] = {16.0};
    const double des3 [1] = {2048.0};
    build_meta_host(g_meta3, 3, 16, base3, 1, des3, 1, 19);

    // META4: D=4, 32 levels, base [8,8,8](pad 8) -> desired [32,32,16](pad 16)
    const double base4[3] = {8.0, 8.0, 8.0};
    const double des4 [3] = {32.0, 32.0, 16.0};
    build_meta_host(g_meta4, 4, 32, base4, 3, des4, 3, 19);

    LvlMeta* dm3 = (LvlMeta*)d_ws;
    LvlMeta* dm4 = dm3 + 16;
    (void)hipMemcpyAsync(dm3, g_meta3, sizeof(g_meta3), hipMemcpyHostToDevice, stream);
    (void)hipMemcpyAsync(dm4, g_meta4, sizeof(g_meta4), hipMemcpyHostToDevice, stream);

    if (n <= 0) return;

    const int block = 256;
    const long long th3 = (long long)n * 16;
    const long long th4 = (long long)n * 32;
    const unsigned g3 = (unsigned)((th3 + block - 1) / block);
    const unsigned g4 = (unsigned)((th4 + block - 1) / block);

    hashgrid_encode<3, 16, 4><<<g3, block, 0, stream>>>(
        (const float4*)xyzt, t3, dm3, out3, n);
    hashgrid_encode<4, 32, 5><<<g4, block, 0, stream>>>(
        (const float4*)xyzt, t4, dm4, out4, n);
}